// MagneticAdaptiveplusGraphConvolution_18983755448438
// MI455X (gfx1250) — compile-verified
//
#include <hip/hip_runtime.h>
#include <hip/hip_bf16.h>
#include <math.h>

#define NN   50000
#define EE   800000
#define FF   64
#define HH   128
#define OUTD 32
#define PROP 3

typedef float v2f __attribute__((ext_vector_type(2)));
typedef float v8f __attribute__((ext_vector_type(8)));

// ---------------------------------------------------------------------------
// Kernel 1: per-edge complex weights  wr = w*cos(q*se), wi = w*sin(q*se)
// ---------------------------------------------------------------------------
__global__ void edge_weights_kernel(const float* __restrict__ ent,
                                    const float* __restrict__ cc,
                                    const float* __restrict__ w,
                                    const float* __restrict__ qp,
                                    const float* __restrict__ enc_w,
                                    const float* __restrict__ enc_b,
                                    float* __restrict__ wr,
                                    float* __restrict__ wi) {
    int e = blockIdx.x * blockDim.x + threadIdx.x;
    if (e >= EE) return;
    float se    = enc_w[0] * ent[e] + enc_w[1] * cc[e] + enc_b[0];
    float phase = qp[0] * se;
    float ww    = w[e];
    wr[e] = ww * cosf(phase);
    wi[e] = ww * sinf(phase);
}

// ---------------------------------------------------------------------------
// Kernel 2: one complex SpMM propagation step.
// One wave32 per edge; lane L owns features {2L, 2L+1} (float2, coalesced).
// Scatter via native fp32 atomics (global_atomic_add_f32).
// ---------------------------------------------------------------------------
__global__ void prop_kernel(const float* __restrict__ prevR,
                            const float* __restrict__ prevI,
                            const float* __restrict__ wr,
                            const float* __restrict__ wi,
                            const int*   __restrict__ row,
                            const int*   __restrict__ col,
                            float* __restrict__ curR,
                            float* __restrict__ curI) {
    unsigned tid  = blockIdx.x * blockDim.x + threadIdx.x;
    unsigned e    = tid >> 5;
    unsigned lane = tid & 31;
    if (e >= EE) return;
    int   c = col[e];
    int   r = row[e];
    float a = wr[e];
    float b = wi[e];
    int f = 2 * lane;
    float2 xr = *(const float2*)(prevR + (size_t)c * FF + f);
    float2 xi = *(const float2*)(prevI + (size_t)c * FF + f);
    float mrx = a * xr.x - b * xi.x;
    float mry = a * xr.y - b * xi.y;
    float mix = b * xr.x + a * xi.x;
    float miy = b * xr.y + a * xi.y;
    float* dR = curR + (size_t)r * FF + f;
    float* dI = curI + (size_t)r * FF + f;
    unsafeAtomicAdd(dR,     mrx);
    unsafeAtomicAdd(dR + 1, mry);
    unsafeAtomicAdd(dI,     mix);
    unsafeAtomicAdd(dI + 1, miy);
}

// ---------------------------------------------------------------------------
// Kernel 3: attention pooling over the 4 propagation snapshots.
// One wave32 per node; lane L owns features {2L, 2L+1}.
// ---------------------------------------------------------------------------
__device__ inline float wave_sum(float v) {
    #pragma unroll
    for (int off = 16; off >= 1; off >>= 1)
        v += __shfl_xor(v, off, 32);
    return v;
}

__global__ void attention_kernel(const float* __restrict__ R0, const float* __restrict__ R1,
                                 const float* __restrict__ R2, const float* __restrict__ R3,
                                 const float* __restrict__ I0, const float* __restrict__ I1,
                                 const float* __restrict__ I2, const float* __restrict__ I3,
                                 const float* __restrict__ ratt_w, const float* __restrict__ ratt_b,
                                 const float* __restrict__ iatt_w, const float* __restrict__ iatt_b,
                                 float* __restrict__ real_x, float* __restrict__ imag_x) {
    unsigned tid  = blockIdx.x * blockDim.x + threadIdx.x;
    unsigned node = tid >> 5;
    unsigned lane = tid & 31;
    if (node >= NN) return;

    const float* Rs[4] = {R0, R1, R2, R3};
    const float* Is[4] = {I0, I1, I2, I3};
    int f = 2 * lane;
    float2 rwv = *(const float2*)(ratt_w + f);
    float2 iwv = *(const float2*)(iatt_w + f);

    float2 xr[4], xi[4];
    float  dr[4], di[4];
    #pragma unroll
    for (int s = 0; s < 4; ++s) {
        xr[s] = *(const float2*)(Rs[s] + (size_t)node * FF + f);
        xi[s] = *(const float2*)(Is[s] + (size_t)node * FF + f);
        dr[s] = wave_sum(xr[s].x * rwv.x + xr[s].y * rwv.y);
        di[s] = wave_sum(xi[s].x * iwv.x + xi[s].y * iwv.y);
    }
    float rb = ratt_b[0], ib = iatt_b[0];
    float er[4], ei[4], sr = 0.f, si = 0.f;
    #pragma unroll
    for (int s = 0; s < 4; ++s) {
        float sgr = 1.f / (1.f + expf(-(dr[s] + rb)));
        float sgi = 1.f / (1.f + expf(-(di[s] + ib)));
        er[s] = expf(sgr); sr += er[s];
        ei[s] = expf(sgi); si += ei[s];
    }
    float2 accr = {0.f, 0.f}, acci = {0.f, 0.f};
    #pragma unroll
    for (int s = 0; s < 4; ++s) {
        float wgr = er[s] / sr;
        float wgi = ei[s] / si;
        accr.x += wgr * xr[s].x; accr.y += wgr * xr[s].y;
        acci.x += wgi * xi[s].x; acci.y += wgi * xi[s].y;
    }
    *(float2*)(real_x + (size_t)node * FF + f) = accr;
    *(float2*)(imag_x + (size_t)node * FF + f) = acci;
}

// ---------------------------------------------------------------------------
// fp32 WMMA helper: accumulate C += X[mBase:mBase+16, 0:KLEN] * W^T tile,
// W row-major [out_d, ldw], using V_WMMA_F32_16X16X4_F32.
// A layout: lane&15 = M row, lane>>4 selects K pair {0,1}/{2,3} -> float2.
// B layout: lane&15 = N col (row of W), same K split -> float2.
// KLEN is compile-time so the whole WMMA chain unrolls.
// ---------------------------------------------------------------------------
template <int KLEN>
__device__ inline v8f wmma_gemm_acc(const float* __restrict__ X, int ldx,
                                    const float* __restrict__ W, int ldw,
                                    int mBase, int nBase, int lane, v8f c) {
    int half = lane >> 4;
    int ml   = lane & 15;
    const float* xp = X + (size_t)(mBase + ml) * ldx + 2 * half;
    const float* wp = W + (size_t)(nBase + ml) * ldw + 2 * half;
    #pragma unroll
    for (int kb = 0; kb < KLEN; kb += 4) {
        v2f a = *(const v2f*)(xp + kb);
        v2f b = *(const v2f*)(wp + kb);
        c = __builtin_amdgcn_wmma_f32_16x16x4_f32(
                /*neg_a=*/false, a, /*neg_b=*/false, b,
                /*c_mod=*/(short)0, c, /*reuse_a=*/false, /*reuse_b=*/false);
    }
    return c;
}

// ---------------------------------------------------------------------------
// Kernel 4: complex linear layer with ComReLU.
//   Zr = Xr @ Wr^T + br ; Zi = Xi @ Wi^T + bi ; mask = (Zr>=0) applied to both
// One wave per 16x16 output tile; wave id in block = output-column tile.
// ---------------------------------------------------------------------------
template <int IN_DIM>
__global__ void complex_linear_kernel(const float* __restrict__ Xr,
                                      const float* __restrict__ Xi,
                                      const float* __restrict__ Wr,
                                      const float* __restrict__ Wi,
                                      const float* __restrict__ br,
                                      const float* __restrict__ bi,
                                      float* __restrict__ Zr,
                                      float* __restrict__ Zi,
                                      int out_dim) {
    int lane  = threadIdx.x & 31;
    int wid   = threadIdx.x >> 5;
    int mBase = blockIdx.x * 16;
    int nBase = wid * 16;
    if (nBase >= out_dim) return;   // uniform across the wave

    v8f cr = {};
    v8f ci = {};
    cr = wmma_gemm_acc<IN_DIM>(Xr, IN_DIM, Wr, IN_DIM, mBase, nBase, lane, cr);
    ci = wmma_gemm_acc<IN_DIM>(Xi, IN_DIM, Wi, IN_DIM, mBase, nBase, lane, ci);

    int   n    = nBase + (lane & 15);
    int   half = lane >> 4;
    float bre  = br[n];
    float bie  = bi[n];
    #pragma unroll
    for (int v = 0; v < 8; ++v) {
        int   m  = mBase + v + 8 * half;       // C layout: VGPR v -> M=v / v+8
        float zr = cr[v] + bre;
        float zi = ci[v] + bie;
        float mk = zr >= 0.f ? 1.f : 0.f;
        Zr[(size_t)m * out_dim + n] = zr * mk;
        Zi[(size_t)m * out_dim + n] = zi * mk;
    }
}

// ---------------------------------------------------------------------------
// Kernel 5: output layer  out = [Xr|Xi] @ out_w^T + out_b,  out_w [32, 256]
// ---------------------------------------------------------------------------
__global__ void out_linear_kernel(const float* __restrict__ Xr,
                                  const float* __restrict__ Xi,
                                  const float* __restrict__ W,
                                  const float* __restrict__ b,
                                  float* __restrict__ Out) {
    int lane  = threadIdx.x & 31;
    int wid   = threadIdx.x >> 5;   // 2 waves -> 32 output cols
    int mBase = blockIdx.x * 16;
    int nBase = wid * 16;

    v8f c = {};
    c = wmma_gemm_acc<HH>(Xr, HH, W,      2 * HH, mBase, nBase, lane, c);
    c = wmma_gemm_acc<HH>(Xi, HH, W + HH, 2 * HH, mBase, nBase, lane, c);

    int   n    = nBase + (lane & 15);
    int   half = lane >> 4;
    float bn   = b[n];
    #pragma unroll
    for (int v = 0; v < 8; ++v) {
        int m = mBase + v + 8 * half;
        Out[(size_t)m * OUTD + n] = c[v] + bn;
    }
}

// ---------------------------------------------------------------------------
extern "C" void kernel_launch(void* const* d_in, const int* in_sizes, int n_in,
                              void* d_out, int out_size, void* d_ws, size_t ws_size,
                              hipStream_t stream) {
    const float* real_feature = (const float*)d_in[0];
    const float* imag_feature = (const float*)d_in[1];
    const float* edge_entropy = (const float*)d_in[2];
    const float* edge_cc      = (const float*)d_in[3];
    const float* edge_w       = (const float*)d_in[4];
    const float* exp_q        = (const float*)d_in[5];
    const float* enc_w        = (const float*)d_in[6];
    const float* enc_b        = (const float*)d_in[7];
    const float* ratt_w       = (const float*)d_in[8];
    const float* ratt_b       = (const float*)d_in[9];
    const float* iatt_w       = (const float*)d_in[10];
    const float* iatt_b       = (const float*)d_in[11];
    const float* rlin0_w      = (const float*)d_in[12];
    const float* rlin0_b      = (const float*)d_in[13];
    const float* rlin1_w      = (const float*)d_in[14];
    const float* rlin1_b      = (const float*)d_in[15];
    const float* ilin0_w      = (const float*)d_in[16];
    const float* ilin0_b      = (const float*)d_in[17];
    const float* ilin1_w      = (const float*)d_in[18];
    const float* ilin1_b      = (const float*)d_in[19];
    const float* out_w        = (const float*)d_in[20];
    const float* out_b        = (const float*)d_in[21];
    const int*   row          = (const int*)d_in[22];
    const int*   col          = (const int*)d_in[23];
    float* out = (float*)d_out;

    // ---- workspace layout (floats) ----
    const size_t NF = (size_t)NN * FF;
    const size_t NH = (size_t)NN * HH;
    float* ws   = (float*)d_ws;
    float* wr   = ws;                         // E
    float* wi   = wr + EE;                    // E
    float* Rbuf = wi + EE;                    // 3 * NF (prop steps 1..3, real)
    float* Ibuf = Rbuf + 3 * NF;              // 3 * NF (imag)
    float* rx   = Ibuf + 3 * NF;              // NF pooled real
    float* ix   = rx + NF;                    // NF pooled imag
    float* zr0  = ix + NF;                    // NH
    float* zi0  = zr0 + NH;                   // NH
    float* zr1  = zi0 + NH;                   // NH
    float* zi1  = zr1 + NH;                   // NH

    // propagation outputs are accumulated with atomics -> zero them each call
    hipMemsetAsync(Rbuf, 0, 6 * NF * sizeof(float), stream);

    // 1) edge weights
    edge_weights_kernel<<<(EE + 255) / 256, 256, 0, stream>>>(
        edge_entropy, edge_cc, edge_w, exp_q, enc_w, enc_b, wr, wi);

    // 2) three complex SpMM propagation steps (wave per edge)
    const float* pR = real_feature;
    const float* pI = imag_feature;
    for (int s = 0; s < PROP; ++s) {
        float* cR = Rbuf + (size_t)s * NF;
        float* cI = Ibuf + (size_t)s * NF;
        prop_kernel<<<(EE * 32 + 255) / 256, 256, 0, stream>>>(
            pR, pI, wr, wi, row, col, cR, cI);
        pR = cR;
        pI = cI;
    }

    // 3) attention pooling (wave per node)
    attention_kernel<<<((size_t)NN * 32 + 255) / 256, 256, 0, stream>>>(
        real_feature, Rbuf, Rbuf + NF, Rbuf + 2 * NF,
        imag_feature, Ibuf, Ibuf + NF, Ibuf + 2 * NF,
        ratt_w, ratt_b, iatt_w, iatt_b, rx, ix);

    // 4) complex MLP via fp32 WMMA (16x16x4), fully unrolled K chains
    dim3 blk0(256);                      // 8 waves -> 128 output cols
    dim3 grd(NN / 16);                   // 3125 row tiles (exact)
    complex_linear_kernel<FF><<<grd, blk0, 0, stream>>>(
        rx, ix, rlin0_w, ilin0_w, rlin0_b, ilin0_b, zr0, zi0, HH);
    complex_linear_kernel<HH><<<grd, blk0, 0, stream>>>(
        zr0, zi0, rlin1_w, ilin1_w, rlin1_b, ilin1_b, zr1, zi1, HH);

    // 5) output layer (2 waves -> 32 cols)
    out_linear_kernel<<<grd, dim3(64), 0, stream>>>(zr1, zi1, out_w, out_b, out);
}